// MambaBlock_67680094650660
// MI455X (gfx1250) — compile-verified
//
#include <hip/hip_runtime.h>
#include <math.h>

// ---------------------------------------------------------------------------
// Mamba block forward for MI455X (gfx1250), fp32 end-to-end.
// GEMMs run on the WMMA pipe via V_WMMA_F32_16X16X4_F32 (fp32 A/B/C),
// double-buffered LDS fed by GLOBAL_LOAD_ASYNC_TO_LDS_B128 (ASYNCcnt),
// 16-WMMA bursts per K chunk per wave.
// ---------------------------------------------------------------------------

#define D_MODEL 512
#define D_STATE 16
#define D_CONV  4
#define D_INNER 1024
#define DT_RANK 32
#define BATCH   2
#define SEQ     2048
#define MROWS   (BATCH * SEQ)      // 4096 rows for all GEMMs

typedef float v2f __attribute__((ext_vector_type(2)));
typedef float v8f __attribute__((ext_vector_type(8)));

// gfx1250 async global->LDS copy (ASYNCcnt-tracked, no VGPR staging).
#if defined(__AMDGCN__) && __has_builtin(__builtin_amdgcn_global_load_async_to_lds_b128)
#define ASYNC_LDS 1
#else
#define ASYNC_LDS 0
#endif

#if ASYNC_LDS
typedef int v4i_t __attribute__((ext_vector_type(4)));
typedef v4i_t __attribute__((address_space(1)))* gv4i_p;   // global int4*
typedef v4i_t __attribute__((address_space(3)))* lv4i_p;   // LDS int4*
#endif

__device__ __forceinline__ void stage16_async(const float* g, float* l)
{
#if ASYNC_LDS
    // (global int4* src, lds int4* dst, imm offset, imm cpol)
    __builtin_amdgcn_global_load_async_to_lds_b128(
        (gv4i_p)(size_t)g,     // int -> AS(1) pointer (drops const, keeps VA)
        (lv4i_p)l,             // generic -> AS(3) addrspacecast
        0, 0);
#else
    (void)g; (void)l;
#endif
}

__device__ __forceinline__ void wait_async0()
{
#if ASYNC_LDS
#if __has_builtin(__builtin_amdgcn_s_wait_asynccnt)
    __builtin_amdgcn_s_wait_asynccnt(0);
#else
    asm volatile("s_wait_asynccnt 0" ::: "memory");
#endif
#endif
}

// ---------------------------------------------------------------------------
// fp32 WMMA GEMM:  C[M,N] = A[M,K] (row stride lda) * W[N,K]^T
//
// 256 threads = 8 waves. Block computes a 64x64 C tile; wave (wm,wn) computes
// a 16x32 strip (two 16x16 accumulators sharing one A fragment). K is chunked
// by 32 through double-buffered LDS (rows padded to 33 floats -> no bank
// conflicts on fragment gathers). Per chunk: async-DMA the next chunk into
// the other LDS buffer, hoist all LDS fragments into registers, then a dense
// burst of 16 WMMAs; s_wait_asynccnt 0 + barrier closes the chunk.
// ---------------------------------------------------------------------------
__global__ __launch_bounds__(256)
void gemm_wmma_f32(const float* __restrict__ A, int lda,
                   const float* __restrict__ W,
                   float* __restrict__ C,
                   int M, int N, int K)
{
    __shared__ float As[2][64][33];
    __shared__ float Ws[2][64][33];

    const int m0   = blockIdx.y * 64;
    const int n0   = blockIdx.x * 64;
    const int tid  = threadIdx.x;
    const int lane = tid & 31;
    const int wave = tid >> 5;            // 0..7
    const int wm16 = (wave >> 1) * 16;    // 0,16,32,48 (M sub-tile)
    const int wn32 = (wave & 1) * 32;     // 0,32       (N sub-tile)
    const int half = lane >> 4;           // 0: K pair {0,1}; 1: K pair {2,3}
    const int lr   = lane & 15;

    // cooperative-load coordinates: 256 threads x 8 floats = one 64x32 tile
    const int r  = tid >> 2;              // 0..63
    const int c0 = (tid & 3) * 8;         // 0,8,16,24

    const float* pa_base = A + (size_t)(m0 + r) * (size_t)lda + c0;
    const float* pw_base = W + (size_t)(n0 + r) * (size_t)K + c0;

    v8f acc0 = {}, acc1 = {};

    // ---- stage chunk 0 into buffer 0 ----
#if ASYNC_LDS
    stage16_async(pa_base,     &As[0][r][c0]);
    stage16_async(pa_base + 4, &As[0][r][c0 + 4]);
    stage16_async(pw_base,     &Ws[0][r][c0]);
    stage16_async(pw_base + 4, &Ws[0][r][c0 + 4]);
    wait_async0();
#else
    {
        float4 a0 = *reinterpret_cast<const float4*>(pa_base);
        float4 a1 = *reinterpret_cast<const float4*>(pa_base + 4);
        float4 w0 = *reinterpret_cast<const float4*>(pw_base);
        float4 w1 = *reinterpret_cast<const float4*>(pw_base + 4);
        As[0][r][c0 + 0] = a0.x; As[0][r][c0 + 1] = a0.y;
        As[0][r][c0 + 2] = a0.z; As[0][r][c0 + 3] = a0.w;
        As[0][r][c0 + 4] = a1.x; As[0][r][c0 + 5] = a1.y;
        As[0][r][c0 + 6] = a1.z; As[0][r][c0 + 7] = a1.w;
        Ws[0][r][c0 + 0] = w0.x; Ws[0][r][c0 + 1] = w0.y;
        Ws[0][r][c0 + 2] = w0.z; Ws[0][r][c0 + 3] = w0.w;
        Ws[0][r][c0 + 4] = w1.x; Ws[0][r][c0 + 5] = w1.y;
        Ws[0][r][c0 + 6] = w1.z; Ws[0][r][c0 + 7] = w1.w;
    }
#endif
    __syncthreads();

    const int nk = K >> 5;
    for (int kc = 0; kc < nk; ++kc) {
        const int cur = kc & 1;
        const bool pf = (kc + 1 < nk);    // uniform across block

#if ASYNC_LDS
        // kick off DMA of the next K chunk into the other buffer; it overlaps
        // the WMMA burst below and is fenced by s_wait_asynccnt + barrier.
        if (pf) {
            const int nxt = cur ^ 1;
            const float* pa = pa_base + (kc + 1) * 32;
            const float* pw = pw_base + (kc + 1) * 32;
            stage16_async(pa,     &As[nxt][r][c0]);
            stage16_async(pa + 4, &As[nxt][r][c0 + 4]);
            stage16_async(pw,     &Ws[nxt][r][c0]);
            stage16_async(pw + 4, &Ws[nxt][r][c0 + 4]);
        }
#else
        float4 a0, a1, w0, w1;
        if (pf) {
            const float* pa = pa_base + (kc + 1) * 32;
            const float* pw = pw_base + (kc + 1) * 32;
            a0 = *reinterpret_cast<const float4*>(pa);
            a1 = *reinterpret_cast<const float4*>(pa + 4);
            w0 = *reinterpret_cast<const float4*>(pw);
            w1 = *reinterpret_cast<const float4*>(pw + 4);
        }
#endif

        // hoist all LDS fragments for this chunk into registers
        v2f af[8], bf0[8], bf1[8];
        #pragma unroll
        for (int s = 0; s < 8; ++s) {
            const int kk = s * 4 + half * 2;
            af[s].x  = As[cur][wm16 + lr][kk];
            af[s].y  = As[cur][wm16 + lr][kk + 1];
            bf0[s].x = Ws[cur][wn32 + lr][kk];
            bf0[s].y = Ws[cur][wn32 + lr][kk + 1];
            bf1[s].x = Ws[cur][wn32 + 16 + lr][kk];
            bf1[s].y = Ws[cur][wn32 + 16 + lr][kk + 1];
        }

        // dense WMMA burst: 16 back-to-back matrix ops
        #pragma unroll
        for (int s = 0; s < 8; ++s) {
            acc0 = __builtin_amdgcn_wmma_f32_16x16x4_f32(
                false, af[s], false, bf0[s], (short)0, acc0, false, false);
            acc1 = __builtin_amdgcn_wmma_f32_16x16x4_f32(
                false, af[s], false, bf1[s], (short)0, acc1, false, false);
        }

#if ASYNC_LDS
        if (pf) wait_async0();
#else
        if (pf) {
            const int nxt = cur ^ 1;
            As[nxt][r][c0 + 0] = a0.x; As[nxt][r][c0 + 1] = a0.y;
            As[nxt][r][c0 + 2] = a0.z; As[nxt][r][c0 + 3] = a0.w;
            As[nxt][r][c0 + 4] = a1.x; As[nxt][r][c0 + 5] = a1.y;
            As[nxt][r][c0 + 6] = a1.z; As[nxt][r][c0 + 7] = a1.w;
            Ws[nxt][r][c0 + 0] = w0.x; Ws[nxt][r][c0 + 1] = w0.y;
            Ws[nxt][r][c0 + 2] = w0.z; Ws[nxt][r][c0 + 3] = w0.w;
            Ws[nxt][r][c0 + 4] = w1.x; Ws[nxt][r][c0 + 5] = w1.y;
            Ws[nxt][r][c0 + 6] = w1.z; Ws[nxt][r][c0 + 7] = w1.w;
        }
#endif
        __syncthreads();
    }

    // C/D layout: VGPR i -> row i (lanes 0-15) / row i+8 (lanes 16-31);
    // col = lane & 15 within each 16x16 sub-tile.
    const int colb  = n0 + wn32 + lr;
    const int rbase = m0 + wm16 + half * 8;
    #pragma unroll
    for (int i = 0; i < 8; ++i) {
        C[(size_t)(rbase + i) * (size_t)N + colb]      = acc0[i];
        C[(size_t)(rbase + i) * (size_t)N + colb + 16] = acc1[i];
    }
}

// ---------------------------------------------------------------------------
// Depthwise causal conv (k=4) over u = xz[..., :D_INNER], then SiLU.
// ---------------------------------------------------------------------------
__global__ void conv_silu_kernel(const float* __restrict__ xz,
                                 const float* __restrict__ cw,
                                 const float* __restrict__ cb,
                                 float* __restrict__ uc)
{
    int idx = blockIdx.x * blockDim.x + threadIdx.x;
    if (idx >= MROWS * D_INNER) return;
    int d = idx % D_INNER;
    int l = (idx / D_INNER) % SEQ;
    int b = idx / (D_INNER * SEQ);

    float acc = cb[d];
    #pragma unroll
    for (int k = 0; k < D_CONV; ++k) {
        int ls = l + k - (D_CONV - 1);
        if (ls >= 0)
            acc += xz[((size_t)(b * SEQ + ls)) * (2 * D_INNER) + d] * cw[d * D_CONV + k];
    }
    uc[idx] = acc / (1.f + __expf(-acc));   // silu
}

// ---------------------------------------------------------------------------
// delta = softplus(gemm_out + dt_proj_b)   (in place)
// ---------------------------------------------------------------------------
__global__ void softplus_bias_kernel(float* __restrict__ delta,
                                     const float* __restrict__ dtb)
{
    int idx = blockIdx.x * blockDim.x + threadIdx.x;
    if (idx >= MROWS * D_INNER) return;
    float v = delta[idx] + dtb[idx % D_INNER];
    delta[idx] = (v > 20.f) ? v : log1pf(__expf(v));
}

// ---------------------------------------------------------------------------
// Selective scan. One thread per (b, d) channel; 16-state recurrence over L
// in registers. Fuses  y = scan + u*D  and  y *= silu(z).
// ---------------------------------------------------------------------------
__global__ __launch_bounds__(256)
void scan_kernel(const float* __restrict__ delta,   // (B,L,D_INNER)
                 const float* __restrict__ uc,      // (B,L,D_INNER)
                 const float* __restrict__ proj,    // (B,L,64): [dt|Bssm|Cssm]
                 const float* __restrict__ xz,      // (B,L,2*D_INNER), z at +D_INNER
                 const float* __restrict__ A_log,   // (D_INNER,16)
                 const float* __restrict__ Dp,      // (D_INNER)
                 float* __restrict__ ygated)        // (B,L,D_INNER)
{
    int t = blockIdx.x * blockDim.x + threadIdx.x;
    if (t >= BATCH * D_INNER) return;
    int b = t / D_INNER;
    int d = t % D_INNER;

    float Ar[D_STATE];
    #pragma unroll
    for (int n = 0; n < D_STATE; ++n)
        Ar[n] = -__expf(A_log[d * D_STATE + n]);

    float h[D_STATE];
    #pragma unroll
    for (int n = 0; n < D_STATE; ++n) h[n] = 0.f;

    const float dpar = Dp[d];

    for (int l = 0; l < SEQ; ++l) {
        size_t row = (size_t)(b * SEQ + l);
        float dlt = delta[row * D_INNER + d];
        float uu  = uc[row * D_INNER + d];
        float zz  = xz[row * (2 * D_INNER) + D_INNER + d];
        const float* pr = proj + row * (DT_RANK + 2 * D_STATE);

        float y = 0.f;
        #pragma unroll
        for (int n = 0; n < D_STATE; ++n) {
            float Bn = pr[DT_RANK + n];
            float Cn = pr[DT_RANK + D_STATE + n];
            float dA = __expf(dlt * Ar[n]);
            h[n] = dA * h[n] + dlt * Bn * uu;
            y += h[n] * Cn;
        }
        y += uu * dpar;
        float sil = zz / (1.f + __expf(-zz));
        ygated[row * D_INNER + d] = y * sil;
    }
}

// ---------------------------------------------------------------------------
// LayerNorm over last dim (512). One wave32 per row, 16 values per lane,
// __shfl_xor reductions.
// ---------------------------------------------------------------------------
__global__ __launch_bounds__(256)
void layernorm_kernel(const float* __restrict__ y,
                      const float* __restrict__ gw,
                      const float* __restrict__ gb,
                      float* __restrict__ yn)
{
    int row  = blockIdx.x * 8 + (threadIdx.x >> 5);
    int lane = threadIdx.x & 31;
    if (row >= MROWS) return;

    const float* p = y + (size_t)row * D_MODEL;
    float v[16];
    float s = 0.f;
    #pragma unroll
    for (int j = 0; j < 16; ++j) { v[j] = p[lane + j * 32]; s += v[j]; }
    #pragma unroll
    for (int off = 16; off >= 1; off >>= 1) s += __shfl_xor(s, off, 32);
    float mu = s / (float)D_MODEL;

    float q = 0.f;
    #pragma unroll
    for (int j = 0; j < 16; ++j) { float dv = v[j] - mu; q += dv * dv; }
    #pragma unroll
    for (int off = 16; off >= 1; off >>= 1) q += __shfl_xor(q, off, 32);
    float rs = rsqrtf(q / (float)D_MODEL + 1e-5f);

    float* po = yn + (size_t)row * D_MODEL;
    #pragma unroll
    for (int j = 0; j < 16; ++j) {
        int c = lane + j * 32;
        po[c] = (v[j] - mu) * rs * gw[c] + gb[c];
    }
}

// ---------------------------------------------------------------------------
// out = gelu(lin_out + lin_b) + x      (exact gelu via erff)
// ---------------------------------------------------------------------------
__global__ void gelu_res_kernel(const float* __restrict__ lin_out,
                                const float* __restrict__ lb,
                                const float* __restrict__ x,
                                float* __restrict__ out)
{
    int idx = blockIdx.x * blockDim.x + threadIdx.x;
    if (idx >= MROWS * D_MODEL) return;
    float v = lin_out[idx] + lb[idx % D_MODEL];
    float g = 0.5f * v * (1.f + erff(v * 0.70710678118654752f));
    out[idx] = g + x[idx];
}

// ---------------------------------------------------------------------------
// Host-side orchestration
// ---------------------------------------------------------------------------
extern "C" void kernel_launch(void* const* d_in, const int* in_sizes, int n_in,
                              void* d_out, int out_size, void* d_ws, size_t ws_size,
                              hipStream_t stream)
{
    (void)in_sizes; (void)n_in; (void)out_size; (void)ws_size;

    const float* x        = (const float*)d_in[0];   // (B,L,512)
    const float* in_w     = (const float*)d_in[1];   // (2048,512)
    const float* conv_w   = (const float*)d_in[2];   // (1024,4)
    const float* conv_b   = (const float*)d_in[3];   // (1024)
    const float* xproj_w  = (const float*)d_in[4];   // (64,1024)
    const float* dtproj_w = (const float*)d_in[5];   // (1024,32)
    const float* dtproj_b = (const float*)d_in[6];   // (1024)
    const float* A_log    = (const float*)d_in[7];   // (1024,16)
    const float* D_param  = (const float*)d_in[8];   // (1024)
    const float* outp_w   = (const float*)d_in[9];   // (512,1024)
    const float* norm_w   = (const float*)d_in[10];  // (512)
    const float* norm_b   = (const float*)d_in[11];  // (512)
    const float* lin_w    = (const float*)d_in[12];  // (512,512)
    const float* lin_b    = (const float*)d_in[13];  // (512)
    float* out            = (float*)d_out;           // (B,L,512)

    // workspace layout (floats)
    float* ws      = (float*)d_ws;
    float* xz      = ws;                                      // 4096*2048
    float* u_conv  = xz      + (size_t)MROWS * 2 * D_INNER;   // 4096*1024
    float* proj    = u_conv  + (size_t)MROWS * D_INNER;       // 4096*64
    float* delta   = proj    + (size_t)MROWS * 64;            // 4096*1024
    float* ygated  = delta   + (size_t)MROWS * D_INNER;       // 4096*1024
    float* y2      = ygated  + (size_t)MROWS * D_INNER;       // 4096*512
    float* yn      = y2      + (size_t)MROWS * D_MODEL;       // 4096*512
    float* lin_out = yn      + (size_t)MROWS * D_MODEL;       // 4096*512

    dim3 blk256(256);

    // 1) xz = x @ in_proj_w^T           (4096 x 2048 x 512)
    gemm_wmma_f32<<<dim3(2 * D_INNER / 64, MROWS / 64), blk256, 0, stream>>>(
        x, D_MODEL, in_w, xz, MROWS, 2 * D_INNER, D_MODEL);

    // 2) depthwise conv + SiLU -> u_conv
    {
        int n = MROWS * D_INNER;
        conv_silu_kernel<<<(n + 255) / 256, blk256, 0, stream>>>(xz, conv_w, conv_b, u_conv);
    }

    // 3) proj = u_conv @ x_proj_w^T     (4096 x 64 x 1024)
    gemm_wmma_f32<<<dim3(64 / 64, MROWS / 64), blk256, 0, stream>>>(
        u_conv, D_INNER, xproj_w, proj, MROWS, 64, D_INNER);

    // 4) delta_raw = dt @ dt_proj_w^T   (4096 x 1024 x 32); dt = proj[:, :32] (lda=64)
    gemm_wmma_f32<<<dim3(D_INNER / 64, MROWS / 64), blk256, 0, stream>>>(
        proj, 64, dtproj_w, delta, MROWS, D_INNER, DT_RANK);

    // 5) delta = softplus(delta_raw + dt_proj_b)
    {
        int n = MROWS * D_INNER;
        softplus_bias_kernel<<<(n + 255) / 256, blk256, 0, stream>>>(delta, dtproj_b);
    }

    // 6) selective scan (+ u*D, * silu(z)) -> ygated
    scan_kernel<<<(BATCH * D_INNER) / 256, blk256, 0, stream>>>(
        delta, u_conv, proj, xz, A_log, D_param, ygated);

    // 7) y2 = ygated @ out_proj_w^T     (4096 x 512 x 1024)
    gemm_wmma_f32<<<dim3(D_MODEL / 64, MROWS / 64), blk256, 0, stream>>>(
        ygated, D_INNER, outp_w, y2, MROWS, D_MODEL, D_INNER);

    // 8) LayerNorm -> yn
    layernorm_kernel<<<MROWS / 8, blk256, 0, stream>>>(y2, norm_w, norm_b, yn);

    // 9) lin_out = yn @ lin_w^T         (4096 x 512 x 512)
    gemm_wmma_f32<<<dim3(D_MODEL / 64, MROWS / 64), blk256, 0, stream>>>(
        yn, D_MODEL, lin_w, lin_out, MROWS, D_MODEL, D_MODEL);

    // 10) out = gelu(lin_out + lin_b) + x
    {
        int n = MROWS * D_MODEL;
        gelu_res_kernel<<<(n + 255) / 256, blk256, 0, stream>>>(lin_out, lin_b, x, out);
    }
}